// SSMEncoder_9603546873867
// MI455X (gfx1250) — compile-verified
//
#include <hip/hip_runtime.h>
#include <hip/hip_bf16.h>

// SSM long-convolution encoder for MI455X (gfx1250, wave32).
// y[b,t,o] = sum_{tau<=t} u[b,tau] * g[t-tau, o],  g = h @ W^T  (projection folded).
// g via f32 WMMA; convolution via 16384-pt LDS FFTs (128KB LDS per WG, CDNA5 has 320KB).

typedef __attribute__((ext_vector_type(2))) float v2f;
typedef __attribute__((ext_vector_type(8))) float v8f;

#define L_SEQ   8192
#define N_DIM   256
#define D_OUT   256
#define B_SZ    8
#define FFT_N   16384
#define FFT_LG  14
#define NTHR    256

// ---------------------------------------------------------------------------
// Kernel 1: g[l][o] = sum_n h[l][n] * W[o][n]   (8192x256 = h(8192x256) @ W^T)
// One 16x16 output tile per wave via V_WMMA_F32_16X16X4_F32, K-loop over n.
// A layout (16x4 f32): lanes 0-15 -> K=0(v0),K=1(v1); lanes 16-31 -> K=2,K=3.
// B layout (4x16 f32): mirrored with N across lanes. D: VGPR i -> M=i / M=i+8.
// ---------------------------------------------------------------------------
__global__ __launch_bounds__(NTHR) void k_proj(const float* __restrict__ h,
                                               const float* __restrict__ W,
                                               float* __restrict__ g) {
    const int wave = threadIdx.x >> 5;
    const int lane = threadIdx.x & 31;
    const int tile = blockIdx.x * 8 + wave;   // 0 .. 8191  (512 l-tiles x 16 o-tiles)
    const int ot = tile & 15;
    const int lt = tile >> 4;
    const int l0 = lt << 4;
    const int o0 = ot << 4;
    const int lh = lane & 15;   // row within tile (A: M, B: N)
    const int kh = lane >> 4;   // K half-select (0 -> K=0,1 ; 1 -> K=2,3)

    const float* __restrict__ hrow = h + (size_t)(l0 + lh) * N_DIM + 2 * kh;
    const float* __restrict__ wrow = W + (size_t)(o0 + lh) * N_DIM + 2 * kh;

    v8f c = {0.f, 0.f, 0.f, 0.f, 0.f, 0.f, 0.f, 0.f};
#pragma unroll 8
    for (int n0 = 0; n0 < N_DIM; n0 += 4) {
        v2f a = *reinterpret_cast<const v2f*>(hrow + n0);
        v2f b = *reinterpret_cast<const v2f*>(wrow + n0);
        c = __builtin_amdgcn_wmma_f32_16x16x4_f32(
                /*neg_a=*/false, a, /*neg_b=*/false, b,
                /*c_mod=*/(short)0, c, /*reuse_a=*/false, /*reuse_b=*/false);
    }
#pragma unroll
    for (int i = 0; i < 8; ++i) {
        g[(size_t)(l0 + i + 8 * kh) * D_OUT + (o0 + lh)] = c[i];
    }
}

// ---------------------------------------------------------------------------
// In-LDS 16384-point radix-2 DIT FFT (input bit-reversed, output natural).
// sgn = +1 forward (exp(-i...)), sgn = -1 inverse (unscaled).
// ---------------------------------------------------------------------------
__device__ __forceinline__ void fft_lds(float2* x, int tid, float sgn) {
#pragma unroll 1
    for (int len = 2; len <= FFT_N; len <<= 1) {
        const int half = len >> 1;
        const float ang = sgn * (-6.28318530717958647692f) / (float)len;
        for (int t = tid; t < (FFT_N / 2); t += NTHR) {
            const int j = t & (half - 1);
            const int i0 = (t << 1) - j;
            const int i1 = i0 + half;
            float sn, cs;
            __sincosf(ang * (float)j, &sn, &cs);
            const float2 a = x[i0];
            const float2 b = x[i1];
            const float br = b.x * cs - b.y * sn;
            const float bi = b.x * sn + b.y * cs;
            x[i0] = make_float2(a.x + br, a.y + bi);
            x[i1] = make_float2(a.x - br, a.y - bi);
        }
        __syncthreads();
    }
}

// ---------------------------------------------------------------------------
// Kernel 2: forward FFT of each zero-padded filter column: G[o][k], k<16384.
// ---------------------------------------------------------------------------
__global__ __launch_bounds__(NTHR) void k_fft_g(const float* __restrict__ g,
                                                float2* __restrict__ G) {
    __shared__ float2 x[FFT_N];   // 128 KB
    const int o = blockIdx.x;
    const int tid = threadIdx.x;
    for (int l = tid; l < FFT_N; l += NTHR) {
        const int r = __brev((unsigned)l) >> (32 - FFT_LG);
        const float v = (l < L_SEQ) ? g[(size_t)l * D_OUT + o] : 0.0f;
        x[r] = make_float2(v, 0.0f);
    }
    __syncthreads();
    fft_lds(x, tid, 1.0f);
    float2* __restrict__ Go = G + (size_t)o * FFT_N;
    for (int k = tid; k < FFT_N; k += NTHR) Go[k] = x[k];
}

// ---------------------------------------------------------------------------
// Kernel 3: forward FFT of each zero-padded input sequence: U[b][k].
// ---------------------------------------------------------------------------
__global__ __launch_bounds__(NTHR) void k_fft_u(const float* __restrict__ u,
                                                float2* __restrict__ U) {
    __shared__ float2 x[FFT_N];   // 128 KB
    const int b = blockIdx.x;
    const int tid = threadIdx.x;
    const float* __restrict__ ub = u + (size_t)b * L_SEQ;
    for (int l = tid; l < FFT_N; l += NTHR) {
        const int r = __brev((unsigned)l) >> (32 - FFT_LG);
        const float v = (l < L_SEQ) ? ub[l] : 0.0f;
        x[r] = make_float2(v, 0.0f);
    }
    __syncthreads();
    fft_lds(x, tid, 1.0f);
    float2* __restrict__ Ub = U + (size_t)b * FFT_N;
    for (int k = tid; k < FFT_N; k += NTHR) Ub[k] = x[k];
}

// ---------------------------------------------------------------------------
// Kernel 4: per (b,o): pointwise U*G, inverse FFT, emit first 8192 real samples.
// y layout: [B, L, DOUT] -> y[((b*L)+t)*DOUT + o]
// ---------------------------------------------------------------------------
__global__ __launch_bounds__(NTHR) void k_conv(const float2* __restrict__ U,
                                               const float2* __restrict__ G,
                                               float* __restrict__ y) {
    __shared__ float2 x[FFT_N];   // 128 KB
    const int o = blockIdx.x;
    const int b = blockIdx.y;
    const int tid = threadIdx.x;
    const float2* __restrict__ Ub = U + (size_t)b * FFT_N;
    const float2* __restrict__ Go = G + (size_t)o * FFT_N;
    for (int k = tid; k < FFT_N; k += NTHR) {
        const float2 uu = Ub[k];
        const float2 gg = Go[k];
        const float pr = uu.x * gg.x - uu.y * gg.y;
        const float pi = uu.x * gg.y + uu.y * gg.x;
        const int r = __brev((unsigned)k) >> (32 - FFT_LG);
        x[r] = make_float2(pr, pi);
    }
    __syncthreads();
    fft_lds(x, tid, -1.0f);   // inverse (unscaled)
    const float scale = 1.0f / (float)FFT_N;
    float* __restrict__ yb = y + ((size_t)b * L_SEQ) * D_OUT + o;
    for (int t = tid; t < L_SEQ; t += NTHR) {
        yb[(size_t)t * D_OUT] = x[t].x * scale;
    }
}

// ---------------------------------------------------------------------------
// Host launcher. Inputs (setup_inputs order): u [8,8192,1] f32, h [8192,256] f32,
// W [256,256] f32. Output: y [8,8192,256] f32.
// Workspace: g (8 MB) | G (32 MB) | U (1 MB)  = 41 MB.
// ---------------------------------------------------------------------------
extern "C" void kernel_launch(void* const* d_in, const int* in_sizes, int n_in,
                              void* d_out, int out_size, void* d_ws, size_t ws_size,
                              hipStream_t stream) {
    const float* u = (const float*)d_in[0];
    const float* h = (const float*)d_in[1];
    const float* W = (const float*)d_in[2];
    float* y = (float*)d_out;

    float* ws = (float*)d_ws;
    float*  g = ws;                                          // 8192*256 floats
    float2* G = (float2*)(ws + (size_t)L_SEQ * D_OUT);       // 256*16384 cplx
    float2* U = (float2*)(ws + (size_t)L_SEQ * D_OUT
                             + (size_t)D_OUT * FFT_N * 2);   // 8*16384 cplx

    // 1) g = h @ W^T  (WMMA f32): 8192 tiles of 16x16, 8 waves/block.
    k_proj<<<dim3(8192 / 8), dim3(NTHR), 0, stream>>>(h, W, g);
    // 2) G[o] = FFT_16384(pad(g[:,o]))
    k_fft_g<<<dim3(D_OUT), dim3(NTHR), 0, stream>>>(g, G);
    // 3) U[b] = FFT_16384(pad(u[b,:]))
    k_fft_u<<<dim3(B_SZ), dim3(NTHR), 0, stream>>>(u, U);
    // 4) y[b,:,o] = Re(IFFT(U[b] * G[o]))[:8192] / 16384
    k_conv<<<dim3(D_OUT, B_SZ), dim3(NTHR), 0, stream>>>(U, G, y);

    (void)in_sizes; (void)n_in; (void)out_size; (void)ws_size;
}